// Sampler_23897198035285
// MI455X (gfx1250) — compile-verified
//
#include <hip/hip_runtime.h>
#include <hip/hip_bf16.h>
#include <math.h>

// ---------------------------------------------------------------------------
// Fused sampler for MI455X (gfx1250, wave32).
// One 512-thread workgroup per row. Two async-to-LDS streaming passes over
// the row's logits (pass 2 expected to hit the 192MB L2), then small LDS-side
// selection. gumbel is gathered only at <=63 kept positions.
// ---------------------------------------------------------------------------

#define NROWS 256
#define V 128000
#define PLEN 64
#define OLEN 32
#define NSEEN (PLEN + OLEN)   // 96
#define NLP 20                // num_logprobs
#define BLK 512               // threads per block (16 wave32)
#define TILE 2048             // floats per tile = BLK * 4
#define NTILES ((V + TILE - 1) / TILE)  // 63
#define NBINS 2048
#define CAND_MAX 1024
#define CAND_TARGET 256
#define KMAX 63
#define EPS 1e-5f

// Monotone (order-preserving) uint key for float compare / histogram.
__device__ __forceinline__ unsigned fkey(float f) {
  unsigned u = __float_as_uint(f);
  return (u & 0x80000000u) ? ~u : (u | 0x80000000u);
}

// Unique 64-bit sort key: primary = value (desc), secondary = vocab idx (asc).
// Strictly larger key == strictly preferred; no ties possible (idx unique).
__device__ __forceinline__ unsigned long long candKey(float v, int idx) {
  return ((unsigned long long)fkey(v) << 32) | (unsigned)(~idx);
}

// Issue one lane's 16B async global->LDS copy of this tile.
// GVS addressing: saddr(u64 row base) + vaddr(i32 byte offset).
__device__ __forceinline__ void issue_tile_async(const float* rowBase, int tile,
                                                 float* ldsTile, int tid) {
  long e0 = (long)tile * TILE + (long)tid * 4;
  if (e0 > (long)V - 4) e0 = (long)V - 4;          // clamp (dup data masked at consume)
  unsigned byteOff = (unsigned)(e0 * 4);
  unsigned ldsAddr = (unsigned)(size_t)(ldsTile + tid * 4);  // low 32 bits = LDS offset
  asm volatile("global_load_async_to_lds_b128 %0, %1, %2"
               :: "v"(ldsAddr), "v"(byteOff),
                  "s"((unsigned long long)(size_t)rowBase)
               : "memory");
}

__device__ __forceinline__ void wait_async_le1() {
  asm volatile("s_wait_asynccnt 1" ::: "memory");
}
__device__ __forceinline__ void wait_async_0() {
  asm volatile("s_wait_asynccnt 0" ::: "memory");
}

// Block-wide argmax over candidate positions [0,n) by candKey.
// Local strided scan -> 1 LDS store -> wave0 finishes with shuffles.
// 2 internal barriers. Returns winning candidate position (same in all lanes).
__device__ int blockArgBest(const float* v, const int* idx, int n,
                            unsigned long long* redK, int* redP, int tid) {
  unsigned long long bk = 0ull;
  int bp = -1;
  for (int i = tid; i < n; i += BLK) {
    unsigned long long k = candKey(v[i], idx[i]);
    if (k > bk) { bk = k; bp = i; }
  }
  redK[tid] = bk;
  redP[tid] = bp;
  __syncthreads();
  if (tid < 32) {
    for (int j = tid + 32; j < BLK; j += 32) {
      unsigned long long k = redK[j];
      if (k > bk) { bk = k; bp = redP[j]; }
    }
    for (int off = 16; off > 0; off >>= 1) {
      unsigned long long ok = __shfl_down(bk, off, 32);
      int op = __shfl_down(bp, off, 32);
      if (ok > bk) { bk = ok; bp = op; }
    }
    if (tid == 0) redP[0] = bp;
  }
  __syncthreads();
  return redP[0];
}

__global__ __launch_bounds__(BLK) void sampler_fused_kernel(
    const float* __restrict__ logits, const float* __restrict__ temperature,
    const int* __restrict__ top_k, const float* __restrict__ top_p,
    const float* __restrict__ pres, const float* __restrict__ freq,
    const float* __restrict__ rep, const int* __restrict__ prompt,
    const int* __restrict__ outtok, const float* __restrict__ gumbel,
    int* __restrict__ out_sampled, int* __restrict__ out_indices,
    float* __restrict__ out_lps, int* __restrict__ out_ranks) {
  const int row = blockIdx.x;
  const int tid = threadIdx.x;
  const float* rowBase = logits + (size_t)row * V;

  __shared__ float tileBuf[2][TILE];          // async staging (double buffer)
  __shared__ unsigned hist[NBINS];
  __shared__ float candV[CAND_MAX];
  __shared__ int   candI[CAND_MAX];
  __shared__ float adjV[CAND_MAX + NSEEN];
  __shared__ int   adjI[CAND_MAX + NSEEN];
  __shared__ float redF[BLK];
  __shared__ int   redP[BLK];
  __shared__ unsigned long long redK[BLK];
  __shared__ int   seenId[NSEEN];
  __shared__ float seenAdj[NSEEN];
  __shared__ float topAdjV[KMAX + 1];
  __shared__ int   topAdjI[KMAX + 1];
  __shared__ float topRawV[NLP];
  __shared__ int   topRawI[NLP];
  __shared__ unsigned candCount;
  __shared__ int   thrBin;
  __shared__ float s_lse;
  __shared__ float sc_T, sc_tp, sc_pp, sc_fp, sc_rp, sc_tdiv;
  __shared__ int   sc_k;
  __shared__ int   s_sampled;
  __shared__ float s_rawTok;

  // ---- setup: scalars, histogram init, seen-token table -------------------
  if (tid == 0) {
    sc_T = temperature[row];
    sc_k = top_k[row];
    sc_tp = top_p[row];
    sc_pp = pres[row];
    sc_fp = freq[row];
    sc_rp = rep[row];
    sc_tdiv = (sc_T < EPS) ? 1.0f : sc_T;
  }
  for (int i = tid; i < NBINS; i += BLK) hist[i] = 0u;
  if (tid < NSEEN) {
    seenId[tid] = (tid < PLEN) ? prompt[row * PLEN + tid]
                               : outtok[row * OLEN + (tid - PLEN)];
  }
  __syncthreads();

  if (tid < NSEEN) {
    int sid = seenId[tid];
    bool owner = true;
    for (int j = 0; j < tid; ++j)
      if (seenId[j] == sid) { owner = false; break; }
    float adj = -INFINITY;
    if (owner) {
      int ocnt = 0;
      for (int j = PLEN; j < NSEEN; ++j)
        if (seenId[j] == sid) ++ocnt;
      float L = rowBase[sid];
      float y = (L > 0.0f) ? (L / sc_rp) : (L * sc_rp);
      y -= sc_fp * (float)ocnt;
      y -= sc_pp * ((ocnt > 0) ? 1.0f : 0.0f);
      adj = y / sc_tdiv;
    }
    seenAdj[tid] = adj;
  }
  __syncthreads();

  // ---- pass 1: stream row (async->LDS): online logsumexp + histogram ------
  float m = -INFINITY, s = 0.0f;
  issue_tile_async(rowBase, 0, tileBuf[0], tid);
  for (int t = 0; t < NTILES; ++t) {
    if (t + 1 < NTILES) {
      issue_tile_async(rowBase, t + 1, tileBuf[(t + 1) & 1], tid);
      wait_async_le1();
    } else {
      wait_async_0();
    }
    const float* buf = tileBuf[t & 1];
    long base = (long)t * TILE + (long)tid * 4;
#pragma unroll
    for (int j = 0; j < 4; ++j) {
      long gidx = base + j;
      if (gidx < V) {
        float f = buf[tid * 4 + j];
        if (f > m) { s = s * expf(m - f) + 1.0f; m = f; }
        else       { s += expf(f - m); }
        atomicAdd(&hist[fkey(f) >> 21], 1u);
      }
    }
    __syncthreads();  // buffer reuse fence before re-issuing into buf[t&1]
  }

  // ---- logsumexp reduce (redF = running max, candV = running scaled sum) --
  redF[tid] = m;
  candV[tid] = s;
  __syncthreads();
  for (int off = BLK >> 1; off > 0; off >>= 1) {
    if (tid < off) {
      float m1 = redF[tid], s1 = candV[tid];
      float m2 = redF[tid + off], s2 = candV[tid + off];
      float M = fmaxf(m1, m2);
      float sn = (M == -INFINITY) ? 0.0f
                                  : (s1 * expf(m1 - M) + s2 * expf(m2 - M));
      redF[tid] = M;
      candV[tid] = sn;
    }
    __syncthreads();
  }
  if (tid == 0) {
    s_lse = redF[0] + logf(candV[0]);
    candCount = 0u;
  }
  __syncthreads();

  // ---- parallel suffix scan over histogram: find threshold bin ------------
  // chunk sums (4 bins per thread)
  {
    int c0 = tid * 4;
    int cs = (int)(hist[c0] + hist[c0 + 1] + hist[c0 + 2] + hist[c0 + 3]);
    redP[tid] = cs;
    __syncthreads();
    // Hillis-Steele inclusive suffix scan: redP[t] = sum of chunks t..511
    for (int off = 1; off < BLK; off <<= 1) {
      int add = (tid + off < BLK) ? redP[tid + off] : 0;
      __syncthreads();
      redP[tid] += add;
      __syncthreads();
    }
    // unique crossing chunk (suffix counts are non-increasing in tid)
    if (redP[tid] >= CAND_TARGET &&
        (tid == BLK - 1 || redP[tid + 1] < CAND_TARGET)) {
      unsigned cum = (tid == BLK - 1) ? 0u : (unsigned)redP[tid + 1];
      int tb = tid * 4;
      for (int b = tid * 4 + 3; b >= tid * 4; --b) {
        cum += hist[b];
        if (cum >= CAND_TARGET) { tb = b; break; }
      }
      thrBin = tb;
    }
  }
  __syncthreads();
  const int tBin = thrBin;

  // ---- pass 2: stream again (L2-resident), collect above-threshold --------
  issue_tile_async(rowBase, 0, tileBuf[0], tid);
  for (int t = 0; t < NTILES; ++t) {
    if (t + 1 < NTILES) {
      issue_tile_async(rowBase, t + 1, tileBuf[(t + 1) & 1], tid);
      wait_async_le1();
    } else {
      wait_async_0();
    }
    const float* buf = tileBuf[t & 1];
    long base = (long)t * TILE + (long)tid * 4;
#pragma unroll
    for (int j = 0; j < 4; ++j) {
      long gidx = base + j;
      if (gidx < V) {
        float f = buf[tid * 4 + j];
        if ((int)(fkey(f) >> 21) >= tBin) {
          unsigned pos = atomicAdd(&candCount, 1u);
          if (pos < CAND_MAX) {
            candV[pos] = f;
            candI[pos] = (int)gidx;
          }
        }
      }
    }
    __syncthreads();
  }
  const int Cn = (int)min(candCount, (unsigned)CAND_MAX);

  // ---- build penalized candidate list (exclude seen; append adjusted seen)
  const float tdiv = sc_tdiv;
  for (int i = tid; i < Cn; i += BLK) {
    int gi = candI[i];
    bool isSeen = false;
    for (int j = 0; j < NSEEN; ++j)
      if (seenId[j] == gi) { isSeen = true; break; }
    adjV[i] = isSeen ? -INFINITY : (candV[i] / tdiv);
    adjI[i] = gi;
  }
  if (tid < NSEEN) {
    adjV[Cn + tid] = seenAdj[tid];
    adjI[Cn + tid] = seenId[tid];
  }
  // raw-selection scratch copy (tileBuf[0] is free now)
  float* rawSel = tileBuf[0];
  for (int i = tid; i < Cn; i += BLK) rawSel[i] = candV[i];
  __syncthreads();
  const int nAdj = Cn + NSEEN;

  // ---- top-20 raw logprobs (selection on scratch copy) --------------------
  for (int r = 0; r < NLP; ++r) {
    int w = blockArgBest(rawSel, candI, Cn, redK, redP, tid);
    if (tid == 0) {
      if (w < 0) w = 0;
      topRawV[r] = candV[w];
      topRawI[r] = candI[w];
      rawSel[w] = -INFINITY;
    }
    __syncthreads();
  }

  // ---- top-k penalized values (destructive on adjV) -----------------------
  int kc = sc_k;
  kc = (kc < 1) ? 1 : ((kc > KMAX) ? KMAX : kc);
  for (int r = 0; r < kc; ++r) {
    int w = blockArgBest(adjV, adjI, nAdj, redK, redP, tid);
    if (tid == 0) {
      if (w < 0) w = 0;
      topAdjV[r] = adjV[w];
      topAdjI[r] = adjI[w];
      adjV[w] = -INFINITY;
    }
    __syncthreads();
  }

  // ---- parallel gumbel gather at kept candidates --------------------------
  if (tid < kc) redF[tid] = gumbel[(size_t)row * V + topAdjI[tid]];
  __syncthreads();

  // ---- top-p mask + gumbel argmax over kept set (serial, <=63 elems) ------
  if (tid == 0) {
    int greedy = topAdjI[0];
    float xmax = topAdjV[0];
    float sum = 0.0f;
    for (int j = 0; j < kc; ++j) sum += expf(topAdjV[j] - xmax);
    float inv = 1.0f / sum;
    float cum = 0.0f, bestSc = -INFINITY;
    int bestIdx = greedy;
    for (int j = 0; j < kc; ++j) {
      bool keep = (cum < sc_tp);   // exclusive cumsum < top_p
      cum += expf(topAdjV[j] - xmax) * inv;
      if (keep) {
        float scv = topAdjV[j] + redF[j];
        if (scv > bestSc) { bestSc = scv; bestIdx = topAdjI[j]; }
      }
    }
    int sampled = (sc_T < EPS) ? greedy : bestIdx;
    s_sampled = sampled;
    s_rawTok = rowBase[sampled];
  }
  __syncthreads();

  // ---- rank = count(logits >= logit[sampled]) over candidates -------------
  {
    float thr = s_rawTok;
    int cnt = 0;
    for (int i = tid; i < Cn; i += BLK)
      if (candV[i] >= thr) ++cnt;
    redP[tid] = cnt;
    __syncthreads();
    for (int off = BLK >> 1; off > 0; off >>= 1) {
      if (tid < off) redP[tid] += redP[tid + off];
      __syncthreads();
    }
  }

  // ---- outputs ------------------------------------------------------------
  const float lse = s_lse;
  if (tid == 0) {
    out_sampled[row] = s_sampled;
    out_ranks[row] = redP[0];
    out_indices[row * (NLP + 1) + 0] = s_sampled;
    out_lps[row * (NLP + 1) + 0] = s_rawTok - lse;
  }
  if (tid < NLP) {
    out_indices[row * (NLP + 1) + 1 + tid] = topRawI[tid];
    out_lps[row * (NLP + 1) + 1 + tid] = topRawV[tid] - lse;
  }
}

extern "C" void kernel_launch(void* const* d_in, const int* in_sizes, int n_in,
                              void* d_out, int out_size, void* d_ws, size_t ws_size,
                              hipStream_t stream) {
  (void)in_sizes; (void)n_in; (void)out_size; (void)d_ws; (void)ws_size;
  const float* logits      = (const float*)d_in[0];
  const float* temperature = (const float*)d_in[1];
  const int*   top_k       = (const int*)d_in[2];
  const float* top_p       = (const float*)d_in[3];
  const float* pres        = (const float*)d_in[4];
  const float* freq        = (const float*)d_in[5];
  const float* rep         = (const float*)d_in[6];
  const int*   prompt      = (const int*)d_in[7];
  const int*   outtok      = (const int*)d_in[8];
  const float* gumbel      = (const float*)d_in[9];
  // d_in[10] = num_logprobs (scalar 20, hardcoded as NLP)

  // outputs concatenated flat: sampled(i32 N), indices(i32 N*21),
  // lps(f32 N*21), ranks(i32 N)
  int*   outS = (int*)d_out;
  int*   outI = outS + NROWS;
  float* outL = (float*)(outI + NROWS * (NLP + 1));
  int*   outR = (int*)(outL + NROWS * (NLP + 1));

  sampler_fused_kernel<<<NROWS, BLK, 0, stream>>>(
      logits, temperature, top_k, top_p, pres, freq, rep, prompt, outtok,
      gumbel, outS, outI, outL, outR);
}